// Generator_15925738734087
// MI455X (gfx1250) — compile-verified
//
#include <hip/hip_runtime.h>

typedef __attribute__((ext_vector_type(2))) float v2f;
typedef __attribute__((ext_vector_type(8))) float v8f;

static constexpr int Sn  = 64;    // seq len
static constexpr int Bn  = 32;    // batch
static constexpr int Dd  = 586;   // feature dim
static constexpr int De  = 583;   // embedding dim
static constexpr int Ln  = 4;     // BN layers
static constexpr int K1n = 17;    // conv1 kernel
static constexpr int CO1 = 16;    // conv1 out channels
static constexpr int KP  = 20;    // padded per-ci K (17 real + 3 zero)
static constexpr int TP  = 32 * KP; // 640 padded GEMM K per co
static constexpr int LO1 = 570;   // conv1 out length
static constexpr int PL1 = 114;   // after avgpool5
static constexpr int C2n = 8;
static constexpr int PL2 = 38;    // after avgpool3
static constexpr int C3n = 32;
static constexpr int LO3 = 36;
static constexpr float EPSf = 1e-5f;
static constexpr int XPAD = 596;  // x row stride in LDS; max read = 575+19 = 594 < 596

// ---------------- Kernel 1: embed + fold 4 BN layers into per-s affine ----------------
__global__ void __launch_bounds__(256)
k_embed_bn(const int* __restrict__ tok, const float* __restrict__ emb,
           const float* __restrict__ pos, const float* __restrict__ gamma,
           const float* __restrict__ beta, float* __restrict__ X,
           float* __restrict__ P, float* __restrict__ Q) {
  const int s = blockIdx.x;
  const int tid = threadIdx.x;
  const float scale = 24.2074368738204098f; // sqrt(586)
  __shared__ float rs[256], rq[256];
  float sum = 0.f, sumsq = 0.f;
  for (int i = tid; i < Bn * Dd; i += 256) {
    int b = i / Dd, d = i - b * Dd;
    float v;
    if (d < De) v = emb[(long)tok[s * Bn + b] * De + d] * scale;
    else        v = pos[(s * Bn + b) * 3 + (d - De)];
    X[((long)s * Bn + b) * Dd + d] = v;
    sum += v; sumsq += v * v;
  }
  rs[tid] = sum; rq[tid] = sumsq;
  __syncthreads();
  for (int off = 128; off > 0; off >>= 1) {
    if (tid < off) { rs[tid] += rs[tid + off]; rq[tid] += rq[tid + off]; }
    __syncthreads();
  }
  if (tid == 0) {
    const float N = (float)(Bn * Dd);
    float m = rs[0] / N;
    float v = rq[0] / N - m * m;           // biased var, training-mode stats
    float Pv = 1.f, Qv = 0.f;
    for (int l = 0; l < Ln; ++l) {
      float g  = gamma[l * Sn + s];
      float bb = beta[l * Sn + s];
      float a  = g / sqrtf(v + EPSf);
      Qv = a * (Qv - m) + bb;
      Pv = a * Pv;
      m = bb;          // mean after affine BN layer is beta
      v = a * a * v;   // var scales by a^2
    }
    P[s] = Pv; Q[s] = Qv;
  }
}

// ---------------- Kernel 2: conv1 as implicit-im2col GEMM on fp32 WMMA ----------------
// Per s: D[j][co] = sum_t A[j][t] * W[t][co].  K padded to 32*20 with zero weights so
// the K walk is (ci, k0) with constant offsets — no div/mod in the hot loop.
// BN affine folded post-GEMM: y1 = P[s]*D + Q[s]*Wsum[co] + b1[co]
__global__ void __launch_bounds__(256)
k_conv1_wmma(const float* __restrict__ X, const float* __restrict__ w1,
             const float* __restrict__ b1, const float* __restrict__ P,
             const float* __restrict__ Q, float* __restrict__ Y1) {
  const int n = blockIdx.x;
  const int tid = threadIdx.x;
  const int lane = tid & 31;
  const int wave = tid >> 5;

  __shared__ float xl[32 * XPAD];   // 76288 B (zero-padded tail)
  __shared__ float wpad[CO1 * TP];  // 40960 B, [co][ci][20] with k>=17 zeroed
  __shared__ float wsum[CO1];

  for (int i = tid; i < 32 * XPAD; i += 256) {
    int ci = i / XPAD, d = i - ci * XPAD;
    xl[i] = (d < Dd) ? X[((long)n * Bn + ci) * Dd + d] : 0.f;
  }
  for (int i = tid; i < CO1 * TP; i += 256) {
    int co = i / TP, t = i - co * TP;
    int ci = t / KP, k = t - ci * KP;
    wpad[i] = (k < K1n) ? w1[(co * 32 + ci) * K1n + k] : 0.f;
  }
  __syncthreads();
  if (tid < CO1) {
    float ssum = 0.f;
    for (int t = 0; t < TP; ++t) ssum += wpad[tid * TP + t];   // pads are zero
    wsum[tid] = ssum;
  }
  __syncthreads();

  const float Pn = P[n], Qn = Q[n];
  const int rowA = lane & 15;        // A: M index; B: N (co) index
  const int hi   = (lane >> 4) & 1;  // lane group -> K pair {0,1} vs {2,3}

  for (int mt = wave; mt < 36; mt += 8) {
    const int j = mt * 16 + rowA;    // output position for this lane's A row
    v8f acc = {0.f, 0.f, 0.f, 0.f, 0.f, 0.f, 0.f, 0.f};
    for (int ci = 0; ci < 32; ++ci) {
      const float* xp = xl + ci * XPAD + j + 2 * hi;
      const v2f*   wp = (const v2f*)(wpad + rowA * TP + ci * KP + 2 * hi); // 8B aligned
#pragma unroll
      for (int k0 = 0; k0 < 5; ++k0) {
        v2f a, b;
        a.x = xp[k0 * 4];
        a.y = xp[k0 * 4 + 1];
        b   = wp[k0 * 2];
        acc = __builtin_amdgcn_wmma_f32_16x16x4_f32(
            /*neg_a=*/false, a, /*neg_b=*/false, b,
            /*c_mod=*/(short)0, acc, /*reuse_a=*/false, /*reuse_b=*/false);
      }
    }
    // D layout: VGPR rr -> M = rr + 8*hi, N = lane&15.  Each lane holds 8
    // consecutive output positions -> vectorized stores, 8B-aligned base.
    const int co = rowA;
    const float addc = Qn * wsum[co] + b1[co];
    const int jbase = mt * 16 + 8 * hi;
    v2f* yp = (v2f*)(Y1 + ((long)n * CO1 + co) * LO1 + jbase);
    const int nchunk = (mt < 35 || hi == 0) ? 4 : 1;  // tile 35 hi-lanes: only jr 568,569
    for (int cch = 0; cch < nchunk; ++cch) {
      v2f st;
      st.x = Pn * acc[cch * 2]     + addc;
      st.y = Pn * acc[cch * 2 + 1] + addc;
      yp[cch] = st;
    }
  }
}

// ---------------- Kernel 3: pool5 -> conv1x1 -> pool3 -> conv3 -> dense -> argmax ----------------
__global__ void __launch_bounds__(128)
k_tail(const float* __restrict__ Y1, const float* __restrict__ w2, const float* __restrict__ b2,
       const float* __restrict__ w3, const float* __restrict__ b3,
       const float* __restrict__ wlm, const float* __restrict__ bl,
       float* __restrict__ out) {
  const int n = blockIdx.x;
  const int tid = threadIdx.x;
  __shared__ float pl1[CO1 * PL1];
  __shared__ float o2[C2n * PL1];
  __shared__ float p2[C2n * PL2];
  __shared__ float o3[C3n * LO3];
  __shared__ float redv[128];
  __shared__ int   redi[128];

  for (int i = tid; i < CO1 * PL1; i += 128) {           // avgpool k=5
    int c = i / PL1, p = i - c * PL1;
    const float* src = Y1 + ((long)n * CO1 + c) * LO1 + p * 5;
    pl1[i] = (src[0] + src[1] + src[2] + src[3] + src[4]) * (1.f / 5.f);
  }
  __syncthreads();
  for (int i = tid; i < C2n * PL1; i += 128) {           // conv 1x1 (16->8)
    int c2 = i / PL1, p = i - c2 * PL1;
    float sum = b2[c2];
    for (int c1 = 0; c1 < CO1; ++c1) sum += pl1[c1 * PL1 + p] * w2[c2 * CO1 + c1];
    o2[i] = sum;
  }
  __syncthreads();
  for (int i = tid; i < C2n * PL2; i += 128) {           // avgpool k=3
    int c = i / PL2, p = i - c * PL2;
    p2[i] = (o2[c * PL1 + p * 3] + o2[c * PL1 + p * 3 + 1] + o2[c * PL1 + p * 3 + 2]) * (1.f / 3.f);
  }
  __syncthreads();
  for (int i = tid; i < C3n * LO3; i += 128) {           // conv k=3 (8->32)
    int c3 = i / LO3, j = i - c3 * LO3;
    float sum = b3[c3];
    for (int c2 = 0; c2 < C2n; ++c2)
      for (int k = 0; k < 3; ++k)
        sum += p2[c2 * PL2 + j + k] * w3[(c3 * C2n + c2) * 3 + k];
    o3[i] = sum;
  }
  __syncthreads();
  // logits + argmax: thread t -> b = t>>2, scans v in [q*32, q*32+32)
  const int b = tid >> 2, q = tid & 3;
  float best = -INFINITY; int bestv = 0;
  for (int v = q * 32; v < q * 32 + 32; ++v) {
    float sum = bl[v];
    for (int k = 0; k < LO3; ++k) sum += o3[b * LO3 + k] * wlm[v * LO3 + k];
    if (sum > best) { best = sum; bestv = v; }           // first-max wins
  }
  redv[tid] = best; redi[tid] = bestv;
  __syncthreads();
  if (q == 0) {
    for (int qq = 1; qq < 4; ++qq) {
      float cv = redv[tid + qq]; int ci = redi[tid + qq];
      if (cv > best) { best = cv; bestv = ci; }          // ascending q keeps lowest v on ties
    }
    out[n * Bn + b] = (float)bestv;
  }
}

extern "C" void kernel_launch(void* const* d_in, const int* in_sizes, int n_in,
                              void* d_out, int out_size, void* d_ws, size_t ws_size,
                              hipStream_t stream) {
  const int*   tok   = (const int*)  d_in[0];
  const float* emb   = (const float*)d_in[1];
  const float* pos   = (const float*)d_in[2];
  const float* gamma = (const float*)d_in[3];
  const float* beta  = (const float*)d_in[4];
  const float* w1    = (const float*)d_in[5];
  const float* b1    = (const float*)d_in[6];
  const float* w2    = (const float*)d_in[7];
  const float* b2    = (const float*)d_in[8];
  const float* w3    = (const float*)d_in[9];
  const float* b3    = (const float*)d_in[10];
  const float* wlm   = (const float*)d_in[11];
  const float* bl    = (const float*)d_in[12];

  float* ws = (float*)d_ws;
  float* X  = ws;                          // 64*32*586 = 1200128 floats
  float* P  = ws + 1200128;                // 64
  float* Q  = P + 64;                      // 64
  float* Y1 = Q + 64;                      // 64*16*570 = 583680 floats

  k_embed_bn  <<<Sn, 256, 0, stream>>>(tok, emb, pos, gamma, beta, X, P, Q);
  k_conv1_wmma<<<Sn, 256, 0, stream>>>(X, w1, b1, P, Q, Y1);
  k_tail      <<<Sn, 128, 0, stream>>>(Y1, w2, b2, w3, b3, wlm, bl, (float*)d_out);
}